// GCNStacking_84696755077903
// MI455X (gfx1250) — compile-verified
//
#include <hip/hip_runtime.h>
#include <hip/hip_bf16.h>

// Problem constants (from reference): B=8, N=2048, C=64, fp32 throughout.
#define BB 8
#define NN 2048
#define CC 64
#define BNC (BB * NN * CC)          // 1,048,576 floats = 4 MB

// gcn_agg tiling
#define KCHUNK 64
#define LDSTR  68                   // floats; 68 % 64 == 4 -> conflict-free b64 operand reads

typedef float v2f __attribute__((ext_vector_type(2)));
typedef float v8f __attribute__((ext_vector_type(8)));

static __device__ __forceinline__ v8f wmma_f32_k4(v2f a, v2f b, v8f c) {
  // D = A(16x4,f32) * B(4x16,f32) + C(16x16,f32); fp32-exact WMMA on CDNA5.
  return __builtin_amdgcn_wmma_f32_16x16x4_f32(
      /*neg_a=*/false, a, /*neg_b=*/false, b,
      /*c_mod=*/(short)0, c, /*reuse_a=*/false, /*reuse_b=*/false);
}

// Low 32 bits of a generic pointer into __shared__ are the LDS byte address
// (shared aperture is 4GB-aligned).
static __device__ __forceinline__ unsigned lds_off32(const void* p) {
  return (unsigned)(size_t)p;
}

// CDNA5 async copy: global -> LDS, 16B per lane, tracked by ASYNCcnt.
// VDST supplies the per-lane LDS byte address, VADDR the 64-bit global address.
static __device__ __forceinline__ void async_copy_b128(unsigned lds_byte, const float* g) {
  asm volatile("global_load_async_to_lds_b128 %0, %1, off"
               :: "v"(lds_byte), "v"(g) : "memory");
}

static __device__ __forceinline__ void wait_async0() {
#if __has_builtin(__builtin_amdgcn_s_wait_asynccnt)
  __builtin_amdgcn_s_wait_asynccnt(0);
#else
  asm volatile("s_wait_asynccnt 0" ::: "memory");
#endif
}

// ---------------------------------------------------------------------------
// Kernel 1: per-node linear transforms.
//   Mout[b,n,o] = sum_c H[b,n,c] * Wm[o,c]
//   Sout[b,n,o] = sum_c H[b,n,c] * Ws[o,c] + bias[o]
// One wave per 16x16 output tile; K=64 swept in steps of 4 with f32 WMMA.
// ---------------------------------------------------------------------------
__global__ __launch_bounds__(256) void gcn_xform(
    const float* __restrict__ H,
    const float* __restrict__ Wm,
    const float* __restrict__ Ws,
    const float* __restrict__ bias,
    float* __restrict__ Mout,
    float* __restrict__ Sout)
{
  const int lane = threadIdx.x & 31;
  const int wave = threadIdx.x >> 5;
  const int gw   = blockIdx.x * 8 + wave;        // 4096 waves total
  const int tiles_per_b = (NN / 16) * (CC / 16); // 512
  const int b    = gw / tiles_per_b;
  const int t    = gw % tiles_per_b;
  const int row0 = (t >> 2) * 16;                // CC/16 == 4 col tiles
  const int col0 = (t & 3) * 16;
  const int hlf  = lane >> 4;
  const int l16  = lane & 15;

  const float* __restrict__ Hb = H + (size_t)b * NN * CC;

  v8f accM = {};
  v8f accS = {};

  #pragma unroll
  for (int k = 0; k < CC; k += 4) {
    const int ka = k + 2 * hlf;
    v2f a  = *(const v2f*)(Hb + (size_t)(row0 + l16) * CC + ka);
    v2f bm = *(const v2f*)(Wm + (size_t)(col0 + l16) * CC + ka);
    v2f bs = *(const v2f*)(Ws + (size_t)(col0 + l16) * CC + ka);
    accM = wmma_f32_k4(a, bm, accM);
    accS = wmma_f32_k4(a, bs, accS);
  }

  const float bv = bias[col0 + l16];
  float* __restrict__ Mb = Mout + (size_t)b * NN * CC;
  float* __restrict__ Sb = Sout + (size_t)b * NN * CC;
  #pragma unroll
  for (int r = 0; r < 8; ++r) {
    const int rr = row0 + r + 8 * hlf;
    Mb[(size_t)rr * CC + col0 + l16] = accM[r];
    Sb[(size_t)rr * CC + col0 + l16] = accS[r] + bv;
  }
}

// ---------------------------------------------------------------------------
// Kernel 2: neighbor aggregation + epilogue.
//   Out[b,i,c] = relu?( sum_j A[b,i,j] * Min[b,j,c] + S[b,i,c] )
// 64(row) x 64(col) tile per 256-thread block; K staged in chunks of 64.
// A chunk is DOUBLE-BUFFERED and staged via global_load_async_to_lds_b128
// (ASYNCcnt) so the next chunk streams into LDS while WMMAs run on the
// current one. M chunk (tiny, L2-resident) is prefetched into registers
// before the compute loop and transposed into LDS between two barriers.
// LDS: 2*17KB (A) + 17KB (M) = 52KB. Stride 68 (==4 mod 64) makes the
// 32-lane b64 WMMA operand fetches hit all 64 banks.
// ---------------------------------------------------------------------------
__global__ __launch_bounds__(256) void gcn_agg(
    const float* __restrict__ A,      // [B][N][N]
    const float* __restrict__ Min,    // [B][N][C]
    const float* __restrict__ S,      // [B][N][C]
    float* __restrict__ Out,          // [B][N][C]
    int do_relu)
{
  __shared__ float lA[2][64 * LDSTR];
  __shared__ float lM[64 * LDSTR];    // transposed: [c][k]

  const int tid  = threadIdx.x;
  const int lane = tid & 31;
  const int wave = tid >> 5;
  const int b    = blockIdx.x >> 5;          // 32 row-blocks per batch
  const int row0 = (blockIdx.x & 31) * 64;
  const int hlf  = lane >> 4;
  const int l16  = lane & 15;
  const int wr   = wave & 3;                 // row subtile 0..3
  const int wcb  = (wave >> 2) * 32;         // col base 0 or 32
  const int srow = row0 + wr * 16;

  const float* __restrict__ Ab = A   + (size_t)b * NN * NN;
  const float* __restrict__ Mb = Min + (size_t)b * NN * CC;
  const float* __restrict__ Sb = S   + (size_t)b * NN * CC;

  // Init accumulators from the self-term + bias buffer S.
  v8f c0, c1;
  #pragma unroll
  for (int r = 0; r < 8; ++r) {
    const int rr = srow + r + 8 * hlf;
    c0[r] = Sb[(size_t)rr * CC + wcb + l16];
    c1[r] = Sb[(size_t)rr * CC + wcb + 16 + l16];
  }

  const unsigned lA_base = lds_off32(&lA[0][0]);

  // ---- prologue: stage chunk kc=0 into buffer 0 ----
  {
    #pragma unroll
    for (int i = 0; i < 4; ++i) {
      const int idx = tid + i * 256;           // 0..1023 float4 slots
      const int r   = idx >> 4;
      const int q   = idx & 15;
      async_copy_b128(lA_base + (unsigned)((r * LDSTR + q * 4) * 4),
                      Ab + (size_t)(row0 + r) * NN + q * 4);
    }
    float4 m[4];
    #pragma unroll
    for (int i = 0; i < 4; ++i) {
      const int idx = tid + i * 256;
      const int k   = idx >> 4;
      const int c4  = idx & 15;
      m[i] = *(const float4*)(Mb + (size_t)k * CC + c4 * 4);
    }
    #pragma unroll
    for (int i = 0; i < 4; ++i) {
      const int idx = tid + i * 256;
      const int k   = idx >> 4;
      const int c4  = idx & 15;
      lM[(c4 * 4 + 0) * LDSTR + k] = m[i].x;
      lM[(c4 * 4 + 1) * LDSTR + k] = m[i].y;
      lM[(c4 * 4 + 2) * LDSTR + k] = m[i].z;
      lM[(c4 * 4 + 3) * LDSTR + k] = m[i].w;
    }
    wait_async0();
    __syncthreads();
  }

  for (int kc = 0; kc < NN; kc += KCHUNK) {
    const int  cur  = (kc >> 6) & 1;
    const bool more = (kc + KCHUNK) < NN;

    float4 m[4];
    if (more) {
      const int kn = kc + KCHUNK;
      // Fire the async A staging for the next chunk (other buffer).
      #pragma unroll
      for (int i = 0; i < 4; ++i) {
        const int idx = tid + i * 256;
        const int r   = idx >> 4;
        const int q   = idx & 15;
        async_copy_b128(
            lA_base + (unsigned)((((cur ^ 1) * (64 * LDSTR)) + r * LDSTR + q * 4) * 4),
            Ab + (size_t)(row0 + r) * NN + kn + q * 4);
      }
      // Register-prefetch next M chunk; latency hides under the WMMA loop.
      #pragma unroll
      for (int i = 0; i < 4; ++i) {
        const int idx = tid + i * 256;
        const int k   = idx >> 4;
        const int c4  = idx & 15;
        m[i] = *(const float4*)(Mb + (size_t)(kn + k) * CC + c4 * 4);
      }
    }

    // ---- compute on current buffers ----
    const float* la = &lA[cur][0];
    #pragma unroll
    for (int k = 0; k < KCHUNK; k += 4) {
      const int ka = k + 2 * hlf;
      v2f a  = *(const v2f*)(la + (wr * 16 + l16) * LDSTR + ka);
      v2f b0 = *(const v2f*)(&lM[(wcb      + l16) * LDSTR + ka]);
      v2f b1 = *(const v2f*)(&lM[(wcb + 16 + l16) * LDSTR + ka]);
      c0 = wmma_f32_k4(a, b0, c0);
      c1 = wmma_f32_k4(a, b1, c1);
    }
    __syncthreads();   // all waves done reading lM (and lA[cur])

    if (more) {
      #pragma unroll
      for (int i = 0; i < 4; ++i) {
        const int idx = tid + i * 256;
        const int k   = idx >> 4;
        const int c4  = idx & 15;
        lM[(c4 * 4 + 0) * LDSTR + k] = m[i].x;
        lM[(c4 * 4 + 1) * LDSTR + k] = m[i].y;
        lM[(c4 * 4 + 2) * LDSTR + k] = m[i].z;
        lM[(c4 * 4 + 3) * LDSTR + k] = m[i].w;
      }
    }
    wait_async0();     // this wave's async writes into lA[cur^1] landed
    __syncthreads();   // publish lM + lA[cur^1] block-wide
  }

  float* __restrict__ Ob = Out + (size_t)b * NN * CC;
  #pragma unroll
  for (int r = 0; r < 8; ++r) {
    const int rr = srow + r + 8 * hlf;
    float x0 = c0[r];
    float x1 = c1[r];
    if (do_relu) { x0 = fmaxf(x0, 0.0f); x1 = fmaxf(x1, 0.0f); }
    Ob[(size_t)rr * CC + wcb + l16]      = x0;
    Ob[(size_t)rr * CC + wcb + 16 + l16] = x1;
  }
}

// ---------------------------------------------------------------------------
// Host-side orchestration: 3 layers, each = xform then agg, all on `stream`.
// d_in: 0:X 1:A 2:Wm0 3:Ws0 4:b0 5:Wm1 6:Ws1 7:b1 8:Wm2 9:Ws2 10:b2
// d_ws layout (floats): [0,BNC) = M, [BNC,2*BNC) = S, [2*BNC,3*BNC) = H ping.
// ---------------------------------------------------------------------------
extern "C" void kernel_launch(void* const* d_in, const int* in_sizes, int n_in,
                              void* d_out, int out_size, void* d_ws, size_t ws_size,
                              hipStream_t stream) {
  (void)in_sizes; (void)n_in; (void)out_size; (void)ws_size;

  const float* X = (const float*)d_in[0];
  const float* A = (const float*)d_in[1];
  const float* Wm[3] = { (const float*)d_in[2], (const float*)d_in[5], (const float*)d_in[8]  };
  const float* Ws[3] = { (const float*)d_in[3], (const float*)d_in[6], (const float*)d_in[9]  };
  const float* bv[3] = { (const float*)d_in[4], (const float*)d_in[7], (const float*)d_in[10] };

  float* ws   = (float*)d_ws;
  float* Mbuf = ws;
  float* Sbuf = ws + (size_t)BNC;
  float* Hbuf = ws + (size_t)2 * BNC;
  float* out  = (float*)d_out;

  const int xform_blocks = (BB * (NN / 16) * (CC / 16)) / 8; // 512
  const int agg_blocks   = BB * (NN / 64);                   // 256

  const float* Hin = X;
  for (int l = 0; l < 3; ++l) {
    gcn_xform<<<xform_blocks, 256, 0, stream>>>(Hin, Wm[l], Ws[l], bv[l], Mbuf, Sbuf);
    float* Hout = (l == 2) ? out : Hbuf;
    gcn_agg<<<agg_blocks, 256, 0, stream>>>(A, Mbuf, Sbuf, Hout, (l < 2) ? 1 : 0);
    Hin = Hbuf;
  }
}